// TemporalBranch_26482768347553
// MI455X (gfx1250) — compile-verified
//
#include <hip/hip_runtime.h>
#include <hip/hip_bf16.h>

// ---- problem constants (fixed by reference) ----
#define CRr   16
#define HEADS 4
#define DIMc  64
#define BTb   2
#define Hh    128
#define Ww    128
#define HWp   16384          // H*W
#define BCb   128            // BT*DIM  (b' dimension)

typedef float v2f __attribute__((ext_vector_type(2)));
typedef float v8f __attribute__((ext_vector_type(8)));

// ------------------------------------------------------------------
// kernel 0: zero the stats accumulator
// ------------------------------------------------------------------
__global__ void k_zero(float* __restrict__ p, int n) {
    int i = blockIdx.x * 256 + threadIdx.x;
    if (i < n) p[i] = 0.0f;
}

// ------------------------------------------------------------------
// kernel 1: LayerNorm over channels (c=64) per pixel + transpose to
//           x_in layout (b'=bt*64+ci, t, p)
// ------------------------------------------------------------------
__global__ void __launch_bounds__(256) k_ln(const float* __restrict__ x,
                                            const float* __restrict__ lnw,
                                            const float* __restrict__ lnb,
                                            float* __restrict__ xin) {
    __shared__ float sw[64], sb[64];
    int tid = threadIdx.x;
    if (tid < 64) { sw[tid] = lnw[tid]; sb[tid] = lnb[tid]; }
    __syncthreads();

    int idx = blockIdx.x * 256 + tid;        // 32*HW threads total
    int b = idx >> 14;                       // /HW
    int p = idx & (HWp - 1);

    float xv[64];
    float s = 0.f, s2 = 0.f;
#pragma unroll
    for (int ci = 0; ci < 64; ci++) {
        float v = x[((size_t)(b * 64 + ci)) * HWp + p];
        xv[ci] = v; s += v; s2 = fmaf(v, v, s2);
    }
    float mu  = s * (1.0f / 64.0f);
    float var = s2 * (1.0f / 64.0f) - mu * mu;
    float rs  = rsqrtf(var + 1e-5f);
    int bt = b >> 4, t = b & 15;
#pragma unroll
    for (int ci = 0; ci < 64; ci++) {
        float xn = (xv[ci] - mu) * rs * sw[ci] + sb[ci];
        xin[(((size_t)(bt * 64 + ci)) * CRr + t) * HWp + p] = xn;
    }
}

// ------------------------------------------------------------------
// kernel 2: fused qkv 1x1 conv (WMMA f32 16x16x4) + depthwise 3x3 +
//           attention statistics. Stores only v; q,k collapse to
//           24 scalars per (b',head) via shuffle/LDS/atomic reduce.
//   spatial tile 32x4, halo 1 -> padded 34x6 = 204 px (pad to 208)
// ------------------------------------------------------------------
#define TWx 32
#define THy 4
#define PWx 34
#define PHy 6
#define PADN   (PWx * PHy)   // 204 real padded pixels
#define PADN16 208           // rounded to 13 N-tiles of 16
#define XSTR   18            // xs row stride (even -> 8B aligned pairs, no bank conflicts)

__global__ void __launch_bounds__(256) k_qkv(const float* __restrict__ xin,
                                             const float* __restrict__ qkvw,  // (48,16)
                                             const float* __restrict__ dww,   // (48,1,3,3)
                                             float* __restrict__ vbuf,
                                             float* __restrict__ stats) {
    __shared__ __align__(8) float xs[PADN16 * XSTR];   // staged input  [pixel][t]
    __shared__ float su[48 * PADN16];                  // 1x1 output    [o][pixel]
    __shared__ __align__(8) float swq[48 * 16];
    __shared__ float sdw[48 * 9];
    __shared__ float sst[96];                          // 4 heads * (16 S + 4 nq + 4 nk)

    int tid  = threadIdx.x;
    int bp   = blockIdx.y;                    // b' in [0,128)
    int tile = blockIdx.x;                    // 128 tiles: 4 x-tiles, 32 y-tiles
    int tx = (tile & 3) * TWx;
    int ty = (tile >> 2) * THy;

    for (int i = tid; i < 48 * 16; i += 256) swq[i] = qkvw[i];
    for (int i = tid; i < 48 * 9;  i += 256) sdw[i] = dww[i];
    if (tid < 96) sst[tid] = 0.0f;

    // ---- phase 0: stage x tile (16 t-channels, zero halo) into LDS ----
    for (int i = tid; i < PADN16; i += 256) {
        int py = i / PWx, px = i - py * PWx;
        int gy = ty + py - 1, gx = tx + px - 1;
        bool inb = (i < PADN) && (gy >= 0) && (gy < Hh) && (gx >= 0) && (gx < Ww);
        int gp = (gy << 7) + gx;
#pragma unroll
        for (int t = 0; t < 16; t++)
            xs[i * XSTR + t] = inb ? xin[((size_t)bp * CRr + t) * HWp + gp] : 0.0f;
    }
    __syncthreads();

    // ---- phase 1: u[48,208] = Wqkv[48,16] @ X[16,208] via WMMA f32 16x16x4 ----
    int wave = tid >> 5, lane = tid & 31;
    int l = lane & 15, hi = lane >> 4;
    for (int nt = wave; nt < PADN16 / 16; nt += 8) {   // 13 N-tiles over 8 waves
        int n0 = nt * 16;
        v8f acc0 = (v8f)0.0f, acc1 = (v8f)0.0f, acc2 = (v8f)0.0f;
#pragma unroll
        for (int ks = 0; ks < 4; ks++) {               // K = 16 in 4 steps of 4
            int kk = ks * 4 + 2 * hi;                  // lanes 0-15: {kk,kk+1}; 16-31: {kk+2,kk+3}
            v2f b;                                     // B[k,n] = x_t(k) at pixel n
            b.x = xs[(n0 + l) * XSTR + kk];
            b.y = xs[(n0 + l) * XSTR + kk + 1];
            v2f a0, a1, a2;                            // A[m,k] = qkv_w[m,k]
            a0.x = swq[(l)      * 16 + kk]; a0.y = swq[(l)      * 16 + kk + 1];
            a1.x = swq[(16 + l) * 16 + kk]; a1.y = swq[(16 + l) * 16 + kk + 1];
            a2.x = swq[(32 + l) * 16 + kk]; a2.y = swq[(32 + l) * 16 + kk + 1];
            acc0 = __builtin_amdgcn_wmma_f32_16x16x4_f32(false, a0, false, b, (short)0, acc0, false, false);
            acc1 = __builtin_amdgcn_wmma_f32_16x16x4_f32(false, a1, false, b, (short)0, acc1, false, false);
            acc2 = __builtin_amdgcn_wmma_f32_16x16x4_f32(false, a2, false, b, (short)0, acc2, false, false);
        }
#pragma unroll
        for (int r = 0; r < 8; r++) {
            int mrow = r + hi * 8;                     // C/D: VGPR r -> M=r (lo lanes), M=r+8 (hi)
            su[(mrow)      * PADN16 + n0 + l] = acc0[r];
            su[(16 + mrow) * PADN16 + n0 + l] = acc1[r];
            su[(32 + mrow) * PADN16 + n0 + l] = acc2[r];
        }
    }
    __syncthreads();

    // ---- phase 2: depthwise 3x3 + stats; thread owns (head=tid&3) ----
    int head = tid & 3;
    int sub  = tid >> 2;                      // 0..63
    float S[16], nq[4], nk[4];
#pragma unroll
    for (int i = 0; i < 16; i++) S[i] = 0.0f;
#pragma unroll
    for (int i = 0; i < 4; i++) { nq[i] = 0.0f; nk[i] = 0.0f; }

#pragma unroll
    for (int j = 0; j < (TWx * THy) / 64; j++) {     // 2 pixels / thread
        int pidx = sub + j * 64;                     // 0..127 interior
        int iy = pidx >> 5, ix = pidx & 31;
        int gp = ((ty + iy) << 7) + tx + ix;
        float qv[4], kv[4];
#pragma unroll
        for (int cc = 0; cc < 4; cc++) {
            int oq = head * 4 + cc, ok = 16 + head * 4 + cc, ov = 32 + head * 4 + cc;
            float aq = 0.f, ak = 0.f, av = 0.f;
#pragma unroll
            for (int dy = 0; dy < 3; dy++)
#pragma unroll
                for (int dx = 0; dx < 3; dx++) {
                    int off = (iy + dy) * PWx + (ix + dx);
                    aq = fmaf(sdw[oq * 9 + dy * 3 + dx], su[oq * PADN16 + off], aq);
                    ak = fmaf(sdw[ok * 9 + dy * 3 + dx], su[ok * PADN16 + off], ak);
                    av = fmaf(sdw[ov * 9 + dy * 3 + dx], su[ov * PADN16 + off], av);
                }
            qv[cc] = aq; kv[cc] = ak;
            vbuf[((size_t)bp * CRr + head * 4 + cc) * HWp + gp] = av;
        }
#pragma unroll
        for (int cc = 0; cc < 4; cc++) {
            nq[cc] = fmaf(qv[cc], qv[cc], nq[cc]);
            nk[cc] = fmaf(kv[cc], kv[cc], nk[cc]);
#pragma unroll
            for (int dd = 0; dd < 4; dd++)
                S[cc * 4 + dd] = fmaf(qv[cc], kv[dd], S[cc * 4 + dd]);
        }
    }

    // wave32 reduce across lanes sharing the same head (xor 4,8,16)
#pragma unroll
    for (int i = 0; i < 16; i++) {
        S[i] += __shfl_xor(S[i], 4, 32);
        S[i] += __shfl_xor(S[i], 8, 32);
        S[i] += __shfl_xor(S[i], 16, 32);
    }
#pragma unroll
    for (int i = 0; i < 4; i++) {
        nq[i] += __shfl_xor(nq[i], 4, 32); nq[i] += __shfl_xor(nq[i], 8, 32); nq[i] += __shfl_xor(nq[i], 16, 32);
        nk[i] += __shfl_xor(nk[i], 4, 32); nk[i] += __shfl_xor(nk[i], 8, 32); nk[i] += __shfl_xor(nk[i], 16, 32);
    }
    if ((tid & 31) < 4) {
#pragma unroll
        for (int i = 0; i < 16; i++) atomicAdd(&sst[head * 24 + i], S[i]);
#pragma unroll
        for (int i = 0; i < 4; i++) {
            atomicAdd(&sst[head * 24 + 16 + i], nq[i]);
            atomicAdd(&sst[head * 24 + 20 + i], nk[i]);
        }
    }
    __syncthreads();
    if (tid < 96) atomicAdd(&stats[(size_t)bp * 96 + tid], sst[tid]);
}

// ------------------------------------------------------------------
// kernel 3: normalize Gram matrix by l2 norms, scale by temperature,
//           4-wide softmax. One thread per (b', head, c).
// ------------------------------------------------------------------
__global__ void k_softmax(const float* __restrict__ stats,
                          const float* __restrict__ temp,
                          float* __restrict__ attn) {
    int idx = blockIdx.x * 256 + threadIdx.x;      // 128*4*4 = 2048
    if (idx >= BCb * HEADS * 4) return;
    int c = idx & 3, head = (idx >> 2) & 3, bp = idx >> 4;
    const float* st = stats + (size_t)bp * 96 + head * 24;
    float T  = temp[head];
    float qn = fmaxf(sqrtf(st[16 + c]), 1e-12f);
    float sc[4];
#pragma unroll
    for (int d = 0; d < 4; d++) {
        float kn = fmaxf(sqrtf(st[20 + d]), 1e-12f);
        sc[d] = st[c * 4 + d] / (qn * kn) * T;
    }
    float m = fmaxf(fmaxf(sc[0], sc[1]), fmaxf(sc[2], sc[3]));
    float e[4], s = 0.f;
#pragma unroll
    for (int d = 0; d < 4; d++) { e[d] = __expf(sc[d] - m); s += e[d]; }
    float inv = 1.0f / s;
#pragma unroll
    for (int d = 0; d < 4; d++)
        attn[(size_t)bp * 64 + head * 16 + c * 4 + d] = e[d] * inv;
}

// ------------------------------------------------------------------
// kernel 4: out = attn @ v (4 MACs per output element); y2 overlays x_in
// ------------------------------------------------------------------
__global__ void __launch_bounds__(256) k_apply(const float* __restrict__ vbuf,
                                               const float* __restrict__ attn,
                                               float* __restrict__ y2) {
    __shared__ float sa[64];
    int tid = threadIdx.x;
    int bp  = blockIdx.y;
    if (tid < 64) sa[tid] = attn[(size_t)bp * 64 + tid];
    __syncthreads();
    int p = blockIdx.x * 256 + tid;
    float vv[16];
#pragma unroll
    for (int i = 0; i < 16; i++) vv[i] = vbuf[((size_t)bp * CRr + i) * HWp + p];
#pragma unroll
    for (int t = 0; t < 16; t++) {
        int head = t >> 2, c = t & 3;
        float a = 0.f;
#pragma unroll
        for (int d = 0; d < 4; d++)
            a = fmaf(sa[head * 16 + c * 4 + d], vv[head * 4 + d], a);
        y2[((size_t)bp * CRr + t) * HWp + p] = a;
    }
}

// ------------------------------------------------------------------
// kernel 5: proj 1x1 conv as WMMA f32 16x16x4.
//   d_out[(bt*16+t)*64 + o][p] = sum_ci projw[o][ci] * y2[(bt*64+ci)*16+t][p]
//   Each wave: 16 pixels (one N tile) x all 4 M tiles; K=64 in 16 steps.
// ------------------------------------------------------------------
__global__ void __launch_bounds__(256) k_proj(const float* __restrict__ y2,
                                              const float* __restrict__ projw,
                                              float* __restrict__ out) {
    __shared__ __align__(16) float sw[64 * 64];
    int tid = threadIdx.x;
#pragma unroll
    for (int i = 0; i < 16; i++) sw[tid + i * 256] = projw[tid + i * 256];
    __syncthreads();

    int t  = blockIdx.y;
    int bt = blockIdx.z;
    int wave = tid >> 5, lane = tid & 31;
    int l = lane & 15, hi = lane >> 4;
    int n0 = blockIdx.x * 128 + wave * 16;

    // column base for this lane (ci = 0); consecutive ci stride = CR*HW
    const float* ybase = y2 + ((size_t)(bt * 64) * CRr + t) * HWp + n0 + l;

    v8f acc[4];
#pragma unroll
    for (int mt = 0; mt < 4; mt++) acc[mt] = (v8f)0.0f;

#pragma unroll
    for (int ks = 0; ks < 16; ks++) {
        int k0 = ks * 4;
        int kk = k0 + 2 * hi;                 // lanes 0-15 -> K {k0,k0+1}; 16-31 -> {k0+2,k0+3}
        v2f b;
        b.x = ybase[(size_t)kk * (CRr * HWp)];
        b.y = ybase[(size_t)(kk + 1) * (CRr * HWp)];
#pragma unroll
        for (int mt = 0; mt < 4; mt++) {
            int m = mt * 16 + l;
            v2f a;
            a.x = sw[m * 64 + kk];
            a.y = sw[m * 64 + kk + 1];
            acc[mt] = __builtin_amdgcn_wmma_f32_16x16x4_f32(
                /*neg_a=*/false, a, /*neg_b=*/false, b,
                /*c_mod=*/(short)0, acc[mt],
                /*reuse_a=*/false, /*reuse_b=*/false);
        }
    }

    float* obase = out + ((size_t)(bt * CRr + t) * 64) * HWp;
#pragma unroll
    for (int mt = 0; mt < 4; mt++)
#pragma unroll
        for (int r = 0; r < 8; r++) {
            int m = mt * 16 + r + hi * 8;     // C/D: VGPR r -> rows M=r (lo lanes), M=r+8 (hi lanes)
            obase[(size_t)m * HWp + n0 + l] = acc[mt][r];
        }
}

// ------------------------------------------------------------------
// host launcher
// ------------------------------------------------------------------
extern "C" void kernel_launch(void* const* d_in, const int* in_sizes, int n_in,
                              void* d_out, int out_size, void* d_ws, size_t ws_size,
                              hipStream_t stream) {
    const float* x     = (const float*)d_in[0];   // (32,64,128,128)
    const float* lnw   = (const float*)d_in[1];   // (64)
    const float* lnb   = (const float*)d_in[2];   // (64)
    const float* qkvw  = (const float*)d_in[3];   // (48,16)
    const float* dww   = (const float*)d_in[4];   // (48,1,3,3)
    const float* projw = (const float*)d_in[5];   // (64,64)
    const float* temp  = (const float*)d_in[6];   // (4,1,1)
    float* out = (float*)d_out;

    // workspace layout (floats): x_in | v | stats | attn ; y2 overlays x_in
    float* xin   = (float*)d_ws;
    float* vbuf  = xin  + (size_t)BCb * CRr * HWp;       // +33.5M
    float* stats = vbuf + (size_t)BCb * CRr * HWp;       // +33.5M
    float* attn  = stats + (size_t)BCb * HEADS * 24;     // 12288 floats
    float* y2    = xin;                                   // reuse after k_qkv

    k_zero<<<dim3((BCb * HEADS * 24 + 255) / 256), dim3(256), 0, stream>>>(stats, BCb * HEADS * 24);
    k_ln<<<dim3((32 * HWp) / 256), dim3(256), 0, stream>>>(x, lnw, lnb, xin);
    k_qkv<<<dim3(128, BCb), dim3(256), 0, stream>>>(xin, qkvw, dww, vbuf, stats);
    k_softmax<<<dim3((BCb * HEADS * 4 + 255) / 256), dim3(256), 0, stream>>>(stats, temp, attn);
    k_apply<<<dim3(HWp / 256, BCb), dim3(256), 0, stream>>>(vbuf, attn, y2);
    k_proj<<<dim3(HWp / 128, CRr, BTb), dim3(256), 0, stream>>>(y2, projw, out);
}